// GNNClassifier_79207786873560
// MI455X (gfx1250) — compile-verified
//
#include <hip/hip_runtime.h>
#include <hip/hip_bf16.h>

typedef float v2f __attribute__((ext_vector_type(2)));
typedef float v8f __attribute__((ext_vector_type(8)));

constexpr int NNODES = 20000;
constexpr int NEDGES = 320000;
constexpr int DIN    = 256;
constexpr int DHID   = 256;
constexpr int DOUT   = 64;

__global__ void zero_kernel(float* __restrict__ p, int n) {
    int i = blockIdx.x * blockDim.x + threadIdx.x;
    if (i < n) p[i] = 0.0f;
}

// C[M x N] = act(A[M x K]) @ W^T, W is [N x K] row-major (torch Linear), K == 256.
// Each workgroup (8 waves) owns one 64-wide column block: it stages the 64xK W
// block into LDS (64 KB) in k-pair-major layout, then each wave computes one
// 16(M) x 64(N) strip with V_WMMA_F32_16X16X4_F32 (full fp32), B from LDS.
//
// LDS layout: pair p = k>>1 ->  smem[p*128 + n*2 + {0,1}] = {B[2p][n], B[2p+1][n]}
//  - ds_load_b64: lanes 0-15 read consecutive n (8B stride), lanes 16-31 at
//    +512B (half=1) -> conflict-free across all 64 banks.
template <bool RELU>
__global__ __launch_bounds__(256)
void gemm_wmma_f32(const float* __restrict__ A, const float* __restrict__ W,
                   float* __restrict__ C, int M, int N, int K, int mTiles) {
    __shared__ float smem[256 * 64];                  // 64 KB: K(=256) x 64 cols, pair-packed

    const int tid  = threadIdx.x;
    const int lane = tid & 31;
    const int wave = tid >> 5;
    const int n0   = blockIdx.x << 6;                 // 64-wide column block

    // ---- stage W block into LDS (transposed to k-pair-major) ----
    {
        const int n   = tid & 63;                     // column within block
        const int kq0 = tid >> 6;                     // starting k-quad (0..3)
        for (int kq = kq0; kq < (K >> 2); kq += 4) {
            const float4 v = *(const float4*)(W + (size_t)(n0 + n) * K + (kq << 2));
            v2f lo; lo.x = v.x; lo.y = v.y;
            v2f hi; hi.x = v.z; hi.y = v.w;
            *(v2f*)(smem + (size_t)(2 * kq)     * 128 + 2 * n) = lo;
            *(v2f*)(smem + (size_t)(2 * kq + 1) * 128 + 2 * n) = hi;
        }
    }
    __syncthreads();

    const int mt = blockIdx.y * 8 + wave;             // wave-uniform M tile
    if (mt >= mTiles) return;                         // whole-wave exit, EXEC all-1s
    const int m0 = mt << 4;

    const int half = lane >> 4;                       // 0: K {0,1}; 1: K {2,3} of each quad
    const int l15  = lane & 15;

    // A lane view: row (m0 + l15), starting at K offset 2*half
    const float* arow = A + (size_t)(m0 + l15) * K + 2 * half;
    // Prefetch the A row into cache (gfx1250 global_prefetch_b8)
    __builtin_prefetch(arow, 0, 3);
    __builtin_prefetch(arow + 128, 0, 3);

    // B from LDS: base for this lane; tile t adds t*32 floats, k-step adds 256/quad
    const float* bbase = smem + (size_t)half * 128 + 2 * l15;

    v8f acc0 = {}, acc1 = {}, acc2 = {}, acc3 = {};

    #pragma unroll 8
    for (int k0 = 0; k0 < K; k0 += 4) {
        v2f a = *(const v2f*)(arow + k0);
        if (RELU) { a.x = fmaxf(a.x, 0.0f); a.y = fmaxf(a.y, 0.0f); }
        const float* bp = bbase + (size_t)(k0 >> 1) * 128;
        v2f b0 = *(const v2f*)(bp +  0);
        v2f b1 = *(const v2f*)(bp + 32);
        v2f b2 = *(const v2f*)(bp + 64);
        v2f b3 = *(const v2f*)(bp + 96);
        acc0 = __builtin_amdgcn_wmma_f32_16x16x4_f32(false, a, false, b0, (short)0, acc0, false, false);
        acc1 = __builtin_amdgcn_wmma_f32_16x16x4_f32(false, a, false, b1, (short)0, acc1, false, false);
        acc2 = __builtin_amdgcn_wmma_f32_16x16x4_f32(false, a, false, b2, (short)0, acc2, false, false);
        acc3 = __builtin_amdgcn_wmma_f32_16x16x4_f32(false, a, false, b3, (short)0, acc3, false, false);
    }

    // C/D layout: VGPR v -> row (v + 8*half), column l15 within 16-wide tile
    #pragma unroll
    for (int v = 0; v < 8; ++v) {
        float* crow = C + (size_t)(m0 + v + 8 * half) * N + n0 + l15;
        crow[ 0] = acc0[v];
        crow[16] = acc1[v];
        crow[32] = acc2[v];
        crow[48] = acc3[v];
    }
}

// out[dst[e]] += w[e] * X[src[e]]   (edge-parallel, float4 per thread)
__global__ __launch_bounds__(256)
void scatter_kernel(const float* __restrict__ X, const int* __restrict__ src,
                    const int* __restrict__ dst, const float* __restrict__ w,
                    float* __restrict__ out, int E, int D) {
    const int per = D >> 2;                       // float4 chunks per edge row
    int idx = blockIdx.x * blockDim.x + threadIdx.x;
    int e = idx / per;
    if (e >= E) return;
    int c = (idx % per) << 2;
    float we = w[e];
    const float4 x = *(const float4*)(X + (size_t)src[e] * D + c);
    float* o = out + (size_t)dst[e] * D + c;
    atomicAdd(o + 0, x.x * we);
    atomicAdd(o + 1, x.y * we);
    atomicAdd(o + 2, x.z * we);
    atomicAdd(o + 3, x.w * we);
}

extern "C" void kernel_launch(void* const* d_in, const int* in_sizes, int n_in,
                              void* d_out, int out_size, void* d_ws, size_t ws_size,
                              hipStream_t stream) {
    const float* node_feat = (const float*)d_in[0];   // [20000, 256]
    const int*   edge_src  = (const int*)  d_in[1];   // [320000]
    const int*   edge_dst  = (const int*)  d_in[2];   // [320000]
    const float* edge_w    = (const float*)d_in[3];   // [320000]
    const float* W1        = (const float*)d_in[4];   // [256, 256] row-major [out,in]
    const float* W2        = (const float*)d_in[5];   // [64, 256]
    float* out = (float*)d_out;                       // [20000, 64]

    float* y1   = (float*)d_ws;                       // 20000*256
    float* agg1 = y1 + (size_t)NNODES * DHID;         // 20000*256
    float* y2   = agg1 + (size_t)NNODES * DHID;       // 20000*64

    const int mTiles = NNODES / 16;                   // 1250

    // zero accumulation targets (deterministic across graph replays)
    {
        int n = NNODES * DHID;
        zero_kernel<<<(n + 255) / 256, 256, 0, stream>>>(agg1, n);
        int m = NNODES * DOUT;
        zero_kernel<<<(m + 255) / 256, 256, 0, stream>>>(out, m);
    }

    // Layer 1: y1 = node_feat @ W1^T  (GEMM before scatter; they commute)
    {
        dim3 grid(DHID / 64, (mTiles + 7) / 8);       // 4 x 157
        gemm_wmma_f32<false><<<grid, 256, 0, stream>>>(
            node_feat, W1, y1, NNODES, DHID, DIN, mTiles);
    }
    // agg1[dst] += w * y1[src]
    {
        long long t = (long long)NEDGES * (DHID / 4);
        scatter_kernel<<<(int)((t + 255) / 256), 256, 0, stream>>>(
            y1, edge_src, edge_dst, edge_w, agg1, NEDGES, DHID);
    }
    // Layer 2: y2 = relu(agg1) @ W2^T
    {
        dim3 grid(DOUT / 64, (mTiles + 7) / 8);       // 1 x 157
        gemm_wmma_f32<true><<<grid, 256, 0, stream>>>(
            agg1, W2, y2, NNODES, DOUT, DHID, mTiles);
    }
    // out[dst] += w * y2[src]
    {
        long long t = (long long)NEDGES * (DOUT / 4);
        scatter_kernel<<<(int)((t + 255) / 256), 256, 0, stream>>>(
            y2, edge_src, edge_dst, edge_w, out, NEDGES, DOUT);
    }
}